// BatchelorAdj_64458869179023
// MI455X (gfx1250) — compile-verified
//
#include <hip/hip_runtime.h>
#include <hip/hip_bf16.h>
#include <math.h>

// Problem constants (from reference): NX=NY=320, NC=20, NT=25
#define DIM 320
#define NPX (320 * 320)      // 102400 pixels
#define NCOIL 20
#define NFRAME 25
#define KC 16                // K-slice staged in LDS per iteration
#define LDP 20               // padded LDS row stride (dwords): 20*m mod 64 tiles all
                             // banks (conflict-free ds_load_b64) AND 16B-aligns the
                             // 16B async-to-LDS chunks (80*r + 16*q)
#define PANEL (64 * LDP)     // dwords per plane

typedef float v2f __attribute__((ext_vector_type(2)));
typedef float v8f __attribute__((ext_vector_type(8)));

// ---------------------------------------------------------------------------
// WMMA helpers: V_WMMA_F32_16X16X4_F32  (D = A(16x4) * B(4x16) + C(16x16))
// Operand layouts per CDNA5 ISA 7.12.2 (wave32):
//   A 16x4 f32 : lane = M + 16*(K>>1), vgpr = K&1
//   B 4x16 f32 : lane = N + 16*hi, vgpr v -> K = v + 2*hi
//   C/D 16x16  : lane = N + 16*hi, vgpr v -> M = v + 8*hi
// ---------------------------------------------------------------------------
__device__ __forceinline__ v8f wmma4(v2f a, v2f b, v8f c) {
  return __builtin_amdgcn_wmma_f32_16x16x4_f32(
      /*neg_a=*/false, a, /*neg_b=*/false, b,
      /*c_mod=*/(short)0, c, /*reuse_a=*/false, /*reuse_b=*/false);
}

// 4x16 B-operand chunk from row-major M (ld=DIM), k rows [kb,kb+4), cols [nb,nb+16)
__device__ __forceinline__ v2f load_opB(const float* M, int kb, int nb, int lane) {
  const int n  = nb + (lane & 15);
  const int kh = kb + ((lane >> 4) << 1);        // kb+0 or kb+2
  v2f b;
  b.x = M[kh * DIM + n];
  b.y = M[(kh + 1) * DIM + n];
  return b;
}

// 16x4 A-operand chunk from LDS panel L[64][LDP]; subtile s, k offset kk
__device__ __forceinline__ v2f lds_opA(const float (*L)[LDP], int s, int kk, int lane) {
  return *(const v2f*)&L[(s << 4) + (lane & 15)][kk + ((lane >> 4) << 1)];
}

// ---------------------------------------------------------------------------
// Async DMA of one 64 x KC P-panel (re+im) into LDS, bypassing VGPRs.
// GLOBAL_LOAD_ASYNC_TO_LDS_B128 (GV mode): VDST = per-lane LDS byte offset
// (low 32 bits of the generic shared pointer), VADDR = 64-bit global VA.
// Tracked by ASYNCcnt; completion enforced with s_wait_asynccnt before the
// workgroup barrier publishes the panel.
// ---------------------------------------------------------------------------
__device__ __forceinline__ void stage_async(const float* __restrict__ Pre,
                                            const float* __restrict__ Pim,
                                            float* dstR, float* dstI,
                                            int bm, int kb, int tid) {
#pragma unroll
  for (int i = 0; i < 2; ++i) {
    const int idx = tid + (i << 7);            // 0..255 over 2 iters
    const int r = idx >> 2;                    // panel row 0..63
    const int q = (idx & 3) << 2;              // 16B chunk col 0,4,8,12
    const unsigned dR = (unsigned)(uintptr_t)(dstR + r * LDP + q);
    const unsigned dI = (unsigned)(uintptr_t)(dstI + r * LDP + q);
    const unsigned long long gR =
        (unsigned long long)(uintptr_t)(Pre + (bm + r) * DIM + kb + q);
    const unsigned long long gI =
        (unsigned long long)(uintptr_t)(Pim + (bm + r) * DIM + kb + q);
    asm volatile("global_load_async_to_lds_b128 %0, %1, off"
                 :: "v"(dR), "v"(gR) : "memory");
    asm volatile("global_load_async_to_lds_b128 %0, %1, off"
                 :: "v"(dI), "v"(gI) : "memory");
  }
}

__device__ __forceinline__ void wait_async_all() {
  asm volatile("s_wait_asynccnt 0x0" ::: "memory");
}

// ---------------------------------------------------------------------------
// Kernel 0: centered orthonormal inverse-DFT matrix (symmetric)
//   A[m,n] = exp(+2*pi*i*(m-160)(n-160)/320) / sqrt(320)
// ---------------------------------------------------------------------------
__global__ void k_build_dft(float* __restrict__ Ar, float* __restrict__ Ai) {
  int idx = blockIdx.x * blockDim.x + threadIdx.x;
  if (idx >= NPX) return;
  int m = idx / DIM, n = idx % DIM;
  long long prod = (long long)(m - 160) * (long long)(n - 160);
  int r = (int)(prod % DIM);
  if (r < 0) r += DIM;
  float ang = (float)r * 0.019634954084936207f;  // 2*pi/320
  float s, c;
  __sincosf(ang, &s, &c);
  const float scale = 0.05590169943749474f;      // 1/sqrt(320)
  Ar[idx] = c * scale;
  Ai[idx] = s * scale;
}

__global__ void k_zero(float* __restrict__ p, int n) {
  int idx = blockIdx.x * blockDim.x + threadIdx.x;
  if (idx < n) p[idx] = 0.0f;
}

// ---------------------------------------------------------------------------
// Kernel 1: X_c = kspace_c * mask_{c,t}, repacked into planar per-coil tiles
// ---------------------------------------------------------------------------
__global__ void k_maskmul(const float* __restrict__ kre, const float* __restrict__ kim,
                          const float* __restrict__ mask,
                          float* __restrict__ Xr, float* __restrict__ Xi, int t) {
  int idx = blockIdx.x * blockDim.x + threadIdx.x;   // < NC*NPX
  int p = idx / NCOIL;
  int c = idx % NCOIL;
  float mk = mask[idx * NFRAME + t];
  Xr[c * NPX + p] = kre[idx] * mk;
  Xi[c * NPX + p] = kim[idx] * mk;
}

// ---------------------------------------------------------------------------
// Shared GEMM core: block (128 thr = 4 waves) computes a 64x64 tile of the
// complex product P * Q. Double-buffered P-panels (64 x KC, re/im) are DMA'd
// straight into LDS with global_load_async_to_lds_b128 while the current
// slice's 64 v_wmma execute; s_wait_asynccnt + barrier per slice. Each wave
// owns one 16-col strip x 4 M-subtiles (B operand reused across subtiles).
// Accumulators: P1..P4 per subtile; Cr = P1-P2, Ci = P3+P4.
// ---------------------------------------------------------------------------
#define GEMM_CORE(Pre, Pim, Qre, Qim)                                          \
  __shared__ float lP[2][2][PANEL];  /* [buf][re/im][64*LDP] = 20 KB */        \
  const int tid = threadIdx.x, lane = tid & 31, wid = tid >> 5;                \
  const int bm = blockIdx.x * 64;                                              \
  const int jt = blockIdx.y * 64 + wid * 16;                                   \
  const v8f vzero = {0, 0, 0, 0, 0, 0, 0, 0};                                  \
  v8f p1[4], p2[4], p3[4], p4[4];                                              \
  _Pragma("unroll") for (int s = 0; s < 4; ++s) {                              \
    p1[s] = vzero; p2[s] = vzero; p3[s] = vzero; p4[s] = vzero;                \
  }                                                                            \
  stage_async(Pre, Pim, &lP[0][0][0], &lP[0][1][0], bm, 0, tid);               \
  wait_async_all();                                                            \
  __syncthreads();                                                             \
  for (int it = 0; it < DIM / KC; ++it) {                                      \
    const int cur = it & 1;                                                    \
    const int kb = it * KC;                                                    \
    if (kb + KC < DIM) {  /* DMA next slice into ping buffer (WAR safe:  */    \
      stage_async(Pre, Pim, &lP[cur ^ 1][0][0], &lP[cur ^ 1][1][0],            \
                  bm, kb + KC, tid);         /* prev barrier drained reads */  \
      __builtin_prefetch(Qre + (kb + KC) * DIM + jt, 0, 1);                    \
      __builtin_prefetch(Qim + (kb + KC) * DIM + jt, 0, 1);                    \
    }                                                                          \
    const float(*cR)[LDP] = (const float(*)[LDP]) & lP[cur][0][0];             \
    const float(*cI)[LDP] = (const float(*)[LDP]) & lP[cur][1][0];             \
    _Pragma("unroll") for (int ks = 0; ks < KC / 4; ++ks) {                    \
      const int kk = ks << 2;                                                  \
      const v2f br = load_opB(Qre, kb + kk, jt, lane);                         \
      const v2f bi = load_opB(Qim, kb + kk, jt, lane);                         \
      _Pragma("unroll") for (int s = 0; s < 4; ++s) {                          \
        const v2f ar = lds_opA(cR, s, kk, lane);                               \
        const v2f ai = lds_opA(cI, s, kk, lane);                               \
        p1[s] = wmma4(ar, br, p1[s]);                                          \
        p2[s] = wmma4(ai, bi, p2[s]);                                          \
        p3[s] = wmma4(ar, bi, p3[s]);                                          \
        p4[s] = wmma4(ai, br, p4[s]);                                          \
      }                                                                        \
    }                                                                          \
    wait_async_all();                                                          \
    __syncthreads();                                                           \
  }

// ---------------------------------------------------------------------------
// Kernel 2: T_c = A * X_c  (complex, batched over coils)
// grid = (5, 5, NCOIL), block = 128
// ---------------------------------------------------------------------------
__global__ __launch_bounds__(128) void k_gemm1(
    const float* __restrict__ Ar, const float* __restrict__ Ai,
    const float* __restrict__ Xr, const float* __restrict__ Xi,
    float* __restrict__ Tr, float* __restrict__ Ti) {
  const int c = blockIdx.z;
  const float* qr = Xr + c * NPX;
  const float* qi = Xi + c * NPX;
  GEMM_CORE(Ar, Ai, qr, qi)

  float* trp = Tr + c * NPX;
  float* tip = Ti + c * NPX;
  const int col = jt + (lane & 15);
  const int rhi = (lane >> 4) << 3;
#pragma unroll
  for (int s = 0; s < 4; ++s) {
    const v8f tr = p1[s] - p2[s];
    const v8f ti = p3[s] + p4[s];
    const int rb = bm + (s << 4) + rhi;
#pragma unroll
    for (int v = 0; v < 8; ++v) {
      trp[(rb + v) * DIM + col] = tr[v];
      tip[(rb + v) * DIM + col] = ti[v];
    }
  }
}

// ---------------------------------------------------------------------------
// Kernel 3: Y_c = T_c * A (A symmetric -> B chunks are row-major A), then
// frame_t += conj(S_c) .* Y_c via global atomic f32 adds.
// grid = (5, 5, NCOIL), block = 128
// ---------------------------------------------------------------------------
__global__ __launch_bounds__(128) void k_gemm2(
    const float* __restrict__ Ar, const float* __restrict__ Ai,
    const float* __restrict__ Tr, const float* __restrict__ Ti,
    const float* __restrict__ sre, const float* __restrict__ sim,
    float* __restrict__ Fr, float* __restrict__ Fi) {
  const int c = blockIdx.z;
  const float* trp = Tr + c * NPX;
  const float* tip = Ti + c * NPX;
  GEMM_CORE(trp, tip, Ar, Ai)

  const int col = jt + (lane & 15);
  const int rhi = (lane >> 4) << 3;
#pragma unroll
  for (int s = 0; s < 4; ++s) {
    const v8f yr = p1[s] - p2[s];
    const v8f yi = p3[s] + p4[s];
    const int rb = bm + (s << 4) + rhi;
#pragma unroll
    for (int v = 0; v < 8; ++v) {
      const int pix = (rb + v) * DIM + col;
      const int sidx = pix * NCOIL + c;
      float sr = sre[sidx], si = sim[sidx];
      float fr = sr * yr[v] + si * yi[v];   // conj(S) * Y
      float fi = sr * yi[v] - si * yr[v];
      atomicAdd(&Fr[pix], fr);
      atomicAdd(&Fi[pix], fi);
    }
  }
}

// ---------------------------------------------------------------------------
// Kernel 4: bilinear warp (gather) of every frame at (x+u, y+v), summed over t
// ---------------------------------------------------------------------------
__global__ void k_warp(const float* __restrict__ Fr, const float* __restrict__ Fi,
                       const float* __restrict__ flow, float* __restrict__ out) {
  int idx = blockIdx.x * blockDim.x + threadIdx.x;   // < NPX
  if (idx >= NPX) return;
  const int x = idx / DIM, y = idx % DIM;
  float accR = 0.0f, accI = 0.0f;
  const int fb = idx * 2 * NFRAME;   // flow [Nx,Ny,2,Nt]
  for (int t = 0; t < NFRAME; ++t) {
    float u = flow[fb + t];
    float v = flow[fb + NFRAME + t];
    float xs = fminf(fmaxf((float)x + u, 0.0f), (float)(DIM - 1));
    float ys = fminf(fmaxf((float)y + v, 0.0f), (float)(DIM - 1));
    int x0 = (int)floorf(xs);
    int y0 = (int)floorf(ys);
    int x1 = min(x0 + 1, DIM - 1);
    int y1 = min(y0 + 1, DIM - 1);
    float wx = xs - (float)x0;
    float wy = ys - (float)y0;
    float w00 = (1.0f - wx) * (1.0f - wy);
    float w10 = wx * (1.0f - wy);
    float w01 = (1.0f - wx) * wy;
    float w11 = wx * wy;
    const float* fr = Fr + t * NPX;
    const float* fi = Fi + t * NPX;
    const int i00 = x0 * DIM + y0, i10 = x1 * DIM + y0;
    const int i01 = x0 * DIM + y1, i11 = x1 * DIM + y1;
    accR += fr[i00] * w00 + fr[i10] * w10 + fr[i01] * w01 + fr[i11] * w11;
    accI += fi[i00] * w00 + fi[i10] * w10 + fi[i01] * w01 + fi[i11] * w11;
  }
  out[idx] = accR;          // [0] real plane
  out[NPX + idx] = accI;    // [1] imag plane
}

// ---------------------------------------------------------------------------
// Host orchestration. Workspace layout (floats), total ~54.1 MB:
//   Ar[NPX] Ai[NPX] Xr[NC*NPX] Xi[NC*NPX] Tr[NC*NPX] Ti[NC*NPX]
//   Fr[NT*NPX] Fi[NT*NPX]
// ---------------------------------------------------------------------------
extern "C" void kernel_launch(void* const* d_in, const int* in_sizes, int n_in,
                              void* d_out, int out_size, void* d_ws, size_t ws_size,
                              hipStream_t stream) {
  const float* kre  = (const float*)d_in[0];
  const float* kim  = (const float*)d_in[1];
  const float* mask = (const float*)d_in[2];
  const float* sre  = (const float*)d_in[3];
  const float* sim  = (const float*)d_in[4];
  const float* flow = (const float*)d_in[5];
  float* out = (float*)d_out;

  float* ws = (float*)d_ws;
  float* Ar = ws;
  float* Ai = Ar + NPX;
  float* Xr = Ai + NPX;
  float* Xi = Xr + (size_t)NCOIL * NPX;
  float* Tr = Xi + (size_t)NCOIL * NPX;
  float* Ti = Tr + (size_t)NCOIL * NPX;
  float* Fr = Ti + (size_t)NCOIL * NPX;
  float* Fi = Fr + (size_t)NFRAME * NPX;

  k_build_dft<<<NPX / 256, 256, 0, stream>>>(Ar, Ai);
  k_zero<<<(2 * NFRAME * NPX) / 256, 256, 0, stream>>>(Fr, 2 * NFRAME * NPX);

  for (int t = 0; t < NFRAME; ++t) {
    k_maskmul<<<(NCOIL * NPX) / 256, 256, 0, stream>>>(kre, kim, mask, Xr, Xi, t);
    k_gemm1<<<dim3(5, 5, NCOIL), 128, 0, stream>>>(Ar, Ai, Xr, Xi, Tr, Ti);
    k_gemm2<<<dim3(5, 5, NCOIL), 128, 0, stream>>>(Ar, Ai, Tr, Ti, sre, sim,
                                                   Fr + (size_t)t * NPX,
                                                   Fi + (size_t)t * NPX);
  }
  k_warp<<<NPX / 256, 256, 0, stream>>>(Fr, Fi, flow, out);
}